// DynamicMHC_81097572483876
// MI455X (gfx1250) — compile-verified
//
#include <hip/hip_runtime.h>
#include <hip/hip_bf16.h>
#include <math.h>

#define NS        4
#define CDIM      1024
#define WD        4096          // NS * CDIM
#define TAU_INV   10.0f         // 1 / 0.1
#define RMS_EPS   1.1920929e-07f
#define SINK_ITERS 50

typedef __attribute__((ext_vector_type(2))) float v2f;
typedef __attribute__((ext_vector_type(4))) float v4f;
typedef __attribute__((ext_vector_type(8))) float v8f;

// ---------------------------------------------------------------------------
// Kernel 1: per-token routing. One 256-thread workgroup per token.
// float4 (b128) pass over x computing RMS sumsq + 24 projection dots,
// block-reduce in LDS, Sinkhorn + softmaxes on lane 0, then x_pre writeback.
// ---------------------------------------------------------------------------
__global__ __launch_bounds__(256)
void mhc_router_kernel(const float* __restrict__ x,
                       const float* __restrict__ W_res,     // [16,4096]
                       const float* __restrict__ W_pre,     // [4,4096]
                       const float* __restrict__ W_post,    // [4,4096]
                       const float* __restrict__ H_res_base,// [16]
                       const float* __restrict__ H_pre_base,// [4]
                       const float* __restrict__ H_post_base,//[4]
                       const float* __restrict__ gate,      // [1]
                       float* __restrict__ ws_xpre,         // [BT,1024]
                       float* __restrict__ ws_hres,         // [BT,16]
                       float* __restrict__ ws_hpost)        // [BT,4]
{
    const int t   = blockIdx.x;
    const int tid = threadIdx.x;
    const float* __restrict__ xr = x + (size_t)t * WD;

    __shared__ float red[256 * 26];
    __shared__ float hsh[24];   // [0..15] H_res_eff, [16..19] H_pre, [20..23] H_post

    float ss = 0.0f;
    float d[24];
#pragma unroll
    for (int r = 0; r < 24; ++r) d[r] = 0.0f;

    // 4 iterations of float4 chunks: thread covers [tid*4 + it*1024, +4)
    for (int cb = tid * 4; cb < WD; cb += 1024) {
        const v4f xv = *(const v4f*)(xr + cb);
        ss += xv.x * xv.x + xv.y * xv.y + xv.z * xv.z + xv.w * xv.w;
#pragma unroll
        for (int r = 0; r < 16; ++r) {
            const v4f w = *(const v4f*)(W_res + r * WD + cb);
            d[r] += xv.x * w.x + xv.y * w.y + xv.z * w.z + xv.w * w.w;
        }
#pragma unroll
        for (int r = 0; r < 4; ++r) {
            const v4f w = *(const v4f*)(W_pre + r * WD + cb);
            d[16 + r] += xv.x * w.x + xv.y * w.y + xv.z * w.z + xv.w * w.w;
        }
#pragma unroll
        for (int r = 0; r < 4; ++r) {
            const v4f w = *(const v4f*)(W_post + r * WD + cb);
            d[20 + r] += xv.x * w.x + xv.y * w.y + xv.z * w.z + xv.w * w.w;
        }
    }

    red[tid * 26 + 0] = ss;
#pragma unroll
    for (int r = 0; r < 24; ++r) red[tid * 26 + 1 + r] = d[r];
    __syncthreads();

    for (int off = 128; off > 0; off >>= 1) {
        if (tid < off) {
#pragma unroll
            for (int k = 0; k < 25; ++k)
                red[tid * 26 + k] += red[(tid + off) * 26 + k];
        }
        __syncthreads();
    }

    if (tid == 0) {
        const float scale = rsqrtf(red[0] * (1.0f / (float)WD) + RMS_EPS);

        // --- Sinkhorn in log domain on Z = logits / tau ---
        float Z[16];
#pragma unroll
        for (int k = 0; k < 16; ++k)
            Z[k] = (red[1 + k] * scale + H_res_base[k]) * TAU_INV;

        float u[4] = {0.f, 0.f, 0.f, 0.f};
        float v[4] = {0.f, 0.f, 0.f, 0.f};
        for (int it = 0; it < SINK_ITERS; ++it) {
#pragma unroll
            for (int i = 0; i < 4; ++i) {
                float a0 = Z[i*4+0] + v[0], a1 = Z[i*4+1] + v[1];
                float a2 = Z[i*4+2] + v[2], a3 = Z[i*4+3] + v[3];
                float m  = fmaxf(fmaxf(a0, a1), fmaxf(a2, a3));
                u[i] = -(m + logf(expf(a0-m)+expf(a1-m)+expf(a2-m)+expf(a3-m)));
            }
#pragma unroll
            for (int j = 0; j < 4; ++j) {
                float a0 = Z[0*4+j] + u[0], a1 = Z[1*4+j] + u[1];
                float a2 = Z[2*4+j] + u[2], a3 = Z[3*4+j] + u[3];
                float m  = fmaxf(fmaxf(a0, a1), fmaxf(a2, a3));
                v[j] = -(m + logf(expf(a0-m)+expf(a1-m)+expf(a2-m)+expf(a3-m)));
            }
        }

        const float g = 1.0f / (1.0f + expf(-gate[0]));
#pragma unroll
        for (int i = 0; i < 4; ++i)
#pragma unroll
            for (int j = 0; j < 4; ++j) {
                float h  = expf(Z[i*4+j] + u[i] + v[j]);
                hsh[i*4+j] = g * h + ((i == j) ? (1.0f - g) : 0.0f);
            }

        // --- softmax(pre) ---
        {
            float l[4]; float m = -3.0e38f;
#pragma unroll
            for (int r = 0; r < 4; ++r) { l[r] = red[17 + r] * scale + H_pre_base[r]; m = fmaxf(m, l[r]); }
            float s = 0.f;
#pragma unroll
            for (int r = 0; r < 4; ++r) { l[r] = expf(l[r] - m); s += l[r]; }
            float inv = 1.0f / s;
#pragma unroll
            for (int r = 0; r < 4; ++r) hsh[16 + r] = l[r] * inv;
        }
        // --- softmax(post) ---
        {
            float l[4]; float m = -3.0e38f;
#pragma unroll
            for (int r = 0; r < 4; ++r) { l[r] = red[21 + r] * scale + H_post_base[r]; m = fmaxf(m, l[r]); }
            float s = 0.f;
#pragma unroll
            for (int r = 0; r < 4; ++r) { l[r] = expf(l[r] - m); s += l[r]; }
            float inv = 1.0f / s;
#pragma unroll
            for (int r = 0; r < 4; ++r) hsh[20 + r] = l[r] * inv;
        }

#pragma unroll
        for (int k = 0; k < 16; ++k) ws_hres[(size_t)t * 16 + k] = hsh[k];
#pragma unroll
        for (int k = 0; k < 4;  ++k) ws_hpost[(size_t)t * 4 + k] = hsh[20 + k];
    }
    __syncthreads();

    const float hp0 = hsh[16], hp1 = hsh[17], hp2 = hsh[18], hp3 = hsh[19];
    {
        const int cb = tid * 4;   // 256 threads * 4 floats = 1024 channels
        const v4f x0 = *(const v4f*)(xr + 0*CDIM + cb);
        const v4f x1 = *(const v4f*)(xr + 1*CDIM + cb);
        const v4f x2 = *(const v4f*)(xr + 2*CDIM + cb);
        const v4f x3 = *(const v4f*)(xr + 3*CDIM + cb);
        v4f xp;
        xp.x = x0.x*hp0 + x1.x*hp1 + x2.x*hp2 + x3.x*hp3;
        xp.y = x0.y*hp0 + x1.y*hp1 + x2.y*hp2 + x3.y*hp3;
        xp.z = x0.z*hp0 + x1.z*hp1 + x2.z*hp2 + x3.z*hp3;
        xp.w = x0.w*hp0 + x1.w*hp1 + x2.w*hp2 + x3.w*hp3;
        *(v4f*)(ws_xpre + (size_t)t * CDIM + cb) = xp;
    }
}

// ---------------------------------------------------------------------------
// Kernel 2: fused branch GEMM (y = x_pre @ W_branch) via V_WMMA_F32_16X16X4_F32
// + mixing epilogue. One workgroup (8 wave32) per 32 tokens (2 M-tiles);
// wave w owns output columns [w*128, w*128+128) for both M-tiles, so each
// B fragment is reused by 2 WMMAs (halves the B-load : WMMA ratio).
// ---------------------------------------------------------------------------
__global__ __launch_bounds__(256)
void mhc_branch_mix_kernel(const float* __restrict__ x,        // [BT,4096]
                           const float* __restrict__ Wb,       // [1024,1024]
                           const float* __restrict__ ws_xpre,  // [BT,1024]
                           const float* __restrict__ ws_hres,  // [BT,16]
                           const float* __restrict__ ws_hpost, // [BT,4]
                           float* __restrict__ out)            // [BT,4096]
{
    const int tid     = threadIdx.x;
    const int wave    = tid >> 5;
    const int lane    = tid & 31;
    const int half    = lane >> 4;       // 0: lanes 0-15, 1: lanes 16-31
    const int l15     = lane & 15;
    const int tbase   = blockIdx.x * 32; // 2 M-tiles of 16 tokens
    const int colbase = wave * 128;
    const int koff    = half * 2;        // A/B K-sublane offset per ISA layout

    __shared__ float h_lds[32 * 20];     // per token: 16 H_res_eff + 4 H_post
    for (int idx = tid; idx < 32 * 20; idx += 256) {
        const int tl = idx / 20, k = idx % 20;
        h_lds[idx] = (k < 16) ? ws_hres[(size_t)(tbase + tl) * 16 + k]
                              : ws_hpost[(size_t)(tbase + tl) * 4 + (k - 16)];
    }
    __syncthreads();

    v8f acc[2][8];
#pragma unroll
    for (int mt = 0; mt < 2; ++mt)
#pragma unroll
        for (int nt = 0; nt < 8; ++nt) {
            v8f z = {0.f, 0.f, 0.f, 0.f, 0.f, 0.f, 0.f, 0.f};
            acc[mt][nt] = z;
        }

    // A-matrix rows for this lane (one per M-tile):
    // x_pre[token = tbase + mt*16 + l15][k + koff .. +1]
    const float* __restrict__ aptr0 = ws_xpre + (size_t)(tbase + l15) * CDIM;
    const float* __restrict__ aptr1 = aptr0 + (size_t)16 * CDIM;

    for (int k = 0; k < CDIM; k += 4) {
        const v2f av0 = *(const v2f*)(aptr0 + k + koff);  // 8B aligned (k+koff even)
        const v2f av1 = *(const v2f*)(aptr1 + k + koff);
#pragma unroll
        for (int nt = 0; nt < 8; ++nt) {
            const int col = colbase + nt * 16 + l15;      // B N-index for this lane
            v2f bv;
            bv.x = Wb[(size_t)(k + koff)     * CDIM + col];
            bv.y = Wb[(size_t)(k + koff + 1) * CDIM + col];
            acc[0][nt] = __builtin_amdgcn_wmma_f32_16x16x4_f32(
                false, av0, false, bv, (short)0, acc[0][nt], false, false);
            acc[1][nt] = __builtin_amdgcn_wmma_f32_16x16x4_f32(
                false, av1, false, bv, (short)0, acc[1][nt], false, false);
        }
    }

    // Epilogue: out[t, i*C + c] = sum_j Hres[i,j]*x[t, j*C + c] + y[t,c]*Hpost[i]
#pragma unroll
    for (int mt = 0; mt < 2; ++mt) {
#pragma unroll
        for (int nt = 0; nt < 8; ++nt) {
            const int c = colbase + nt * 16 + l15;
#pragma unroll
            for (int r = 0; r < 8; ++r) {
                const int tl = mt * 16 + r + 8 * half;    // C/D layout: M = r (+8 hi lanes)
                const size_t trow = (size_t)(tbase + tl) * WD;
                const float yv = acc[mt][nt][r];
                const float x0 = x[trow + 0*CDIM + c];
                const float x1 = x[trow + 1*CDIM + c];
                const float x2 = x[trow + 2*CDIM + c];
                const float x3 = x[trow + 3*CDIM + c];
                const float* __restrict__ h = &h_lds[tl * 20];
#pragma unroll
                for (int i = 0; i < 4; ++i) {
                    const float o = h[i*4+0]*x0 + h[i*4+1]*x1 + h[i*4+2]*x2 + h[i*4+3]*x3
                                  + yv * h[16 + i];
                    out[trow + i*CDIM + c] = o;
                }
            }
        }
    }
}

// ---------------------------------------------------------------------------
extern "C" void kernel_launch(void* const* d_in, const int* in_sizes, int n_in,
                              void* d_out, int out_size, void* d_ws, size_t ws_size,
                              hipStream_t stream) {
    const float* x          = (const float*)d_in[0];
    const float* W_res      = (const float*)d_in[1];
    const float* W_pre      = (const float*)d_in[2];
    const float* W_post     = (const float*)d_in[3];
    const float* H_res_base = (const float*)d_in[4];
    const float* H_pre_base = (const float*)d_in[5];
    const float* H_post_base= (const float*)d_in[6];
    const float* gate       = (const float*)d_in[7];
    const float* W_branch   = (const float*)d_in[8];
    float* out = (float*)d_out;

    const int BT = in_sizes[0] / WD;   // B*T tokens (8192)

    float* ws       = (float*)d_ws;
    float* ws_xpre  = ws;                                  // BT*1024
    float* ws_hres  = ws_xpre + (size_t)BT * CDIM;         // BT*16
    float* ws_hpost = ws_hres + (size_t)BT * 16;           // BT*4

    mhc_router_kernel<<<BT, 256, 0, stream>>>(
        x, W_res, W_pre, W_post, H_res_base, H_pre_base, H_post_base, gate,
        ws_xpre, ws_hres, ws_hpost);

    mhc_branch_mix_kernel<<<BT / 32, 256, 0, stream>>>(
        x, W_branch, ws_xpre, ws_hres, ws_hpost, out);

    (void)n_in; (void)out_size; (void)ws_size;
}